// EdgeGenerator_9663676416633
// MI455X (gfx1250) — compile-verified
//
#include <hip/hip_runtime.h>
#include <hip/hip_bf16.h>

#define NROWS 8192
#define DIM   512
#define KTOP  16
#define MBLK  32          // actuator rows per block (two 16-row WMMA groups)

typedef float v2f __attribute__((ext_vector_type(2)));
typedef float v8f __attribute__((ext_vector_type(8)));

// ---------------------------------------------------------------------------
// Kernel 1: per-row inverse L2 norm.  One wave32 per row, float4 loads.
// ---------------------------------------------------------------------------
__global__ void __launch_bounds__(256) inv_norm_kernel(const float* __restrict__ X,
                                                       float* __restrict__ out) {
    const int wave = threadIdx.x >> 5;
    const int lane = threadIdx.x & 31;
    const int row  = blockIdx.x * 8 + wave;
    const float4* p = (const float4*)(X + (size_t)row * DIM);
    float s = 0.0f;
#pragma unroll
    for (int i = 0; i < DIM / 4 / 32; ++i) {          // 4 float4 per lane
        float4 v = p[lane + 32 * i];
        s += v.x * v.x + v.y * v.y + v.z * v.z + v.w * v.w;
    }
#pragma unroll
    for (int off = 16; off > 0; off >>= 1) s += __shfl_down(s, off, 32);
    if (lane == 0) out[row] = rsqrtf(s);
}

// ---------------------------------------------------------------------------
// Kernel 2: fused cosine-similarity GEMM (fp32 WMMA) + per-row top-16.
// One block = 32 actuator rows; 8 wave32s split the 512 sensor column tiles.
// Each B fragment is reused by two WMMAs (row groups), halving L2 traffic.
// ---------------------------------------------------------------------------
__global__ void __launch_bounds__(256, 1) simtopk_kernel(
    const float* __restrict__ A, const float* __restrict__ S,
    const float* __restrict__ inv_na, const float* __restrict__ inv_ns,
    int* __restrict__ e_src, int* __restrict__ e_dst, float* __restrict__ wout)
{
    __shared__ float Alds[MBLK * DIM];        // 64 KB: A tile pre-scaled by 1/||a||
    __shared__ float topv[MBLK][KTOP];        // shared per-row top-16 (sim^2)
    __shared__ int   topi[MBLK][KTOP];        // shared per-row top-16 indices
    __shared__ float thr[MBLK];               // per-row current min of list
    __shared__ int   rowlock[MBLK];           // per-row LDS spinlock

    const int tid     = threadIdx.x;
    const int wave    = tid >> 5;
    const int lane    = tid & 31;
    const int half    = lane >> 4;            // wave32 half-wave (C/A layout)
    const int lane16  = lane & 15;
    const int rowbase = blockIdx.x * MBLK;

    // ---- stage A tile into LDS, pre-scaled by inv_na --------------------
    {
        const float4* Ap = (const float4*)(A + (size_t)rowbase * DIM);
        float4*       Lp = (float4*)Alds;
        for (int i = tid; i < MBLK * DIM / 4; i += 256) {
            int r = i >> 7;                   // 128 float4 per row
            float4 v = Ap[i];
            float sc = inv_na[rowbase + r];
            v.x *= sc; v.y *= sc; v.z *= sc; v.w *= sc;
            Lp[i] = v;
        }
    }
    // ---- init shared top-k lists ----------------------------------------
    for (int i = tid; i < MBLK * KTOP; i += 256) {
        ((float*)topv)[i] = -1.0f;
        ((int*)topi)[i]   = 0;
    }
    if (tid < MBLK) { thr[tid] = -1.0f; rowlock[tid] = 0; }
    __syncthreads();

    // A fragments: lane holds matrix row (lane&15); K pair selected by half.
    const float* Arow0 = Alds + lane16 * DIM + 2 * half;            // rows 0..15
    const float* Arow1 = Alds + (16 + lane16) * DIM + 2 * half;     // rows 16..31

    for (int t = wave; t < NROWS / 16; t += 8) {
        const int colbase = t * 16;
        // B fragment: column n = lane16 -> sensor row (colbase+n)
        const float* Srow = S + (size_t)(colbase + lane16) * DIM + 2 * half;

        v8f acc0 = {};
        v8f acc1 = {};
#pragma unroll 4
        for (int kc = 0; kc < DIM; kc += 4) {
            v2f b  = *(const v2f*)(Srow + kc);    // 1 global b64 ...
            v2f a0 = *(const v2f*)(Arow0 + kc);   // ... feeds 2 WMMAs
            v2f a1 = *(const v2f*)(Arow1 + kc);
            acc0 = __builtin_amdgcn_wmma_f32_16x16x4_f32(
                false, a0, false, b, (short)0, acc0, false, false);
            acc1 = __builtin_amdgcn_wmma_f32_16x16x4_f32(
                false, a1, false, b, (short)0, acc1, false, false);
        }

        const float invn = inv_ns[colbase + lane16];
#pragma unroll
        for (int g = 0; g < 2; ++g) {
            v8f acc = g ? acc1 : acc0;
#pragma unroll
            for (int j = 0; j < 8; ++j) {
                float simv = acc[j] * invn;       // cosine similarity
                float v2v  = simv * simv;         // ranked on sim^2 (reference)
                int   rowj = g * 16 + j + 8 * half;
                float cth  = thr[rowj];
                unsigned mm = (unsigned)__ballot(v2v > cth);
                while (mm) {                      // uniform across the wave
                    int src = __ffs(mm) - 1;
                    mm &= mm - 1;
                    float cv = __shfl(v2v, src, 32);
                    if (lane == 0) {
                        int crow = g * 16 + j + 8 * (src >> 4);
                        int cidx = colbase + (src & 15);
                        while (atomicCAS(&rowlock[crow], 0, 1) != 0) { }
                        float mn = topv[crow][0]; int mnk = 0;
#pragma unroll
                        for (int k = 1; k < KTOP; ++k) {
                            float tv = topv[crow][k];
                            if (tv < mn) { mn = tv; mnk = k; }
                        }
                        if (cv > mn) {
                            topv[crow][mnk] = cv;
                            topi[crow][mnk] = cidx;
                            float nm = topv[crow][0];
#pragma unroll
                            for (int k = 1; k < KTOP; ++k)
                                nm = fminf(nm, topv[crow][k]);
                            thr[crow] = nm;
                        }
                        __threadfence_block();
                        atomicExch(&rowlock[crow], 0);
                    }
                }
            }
        }
    }
    __syncthreads();

    // ---- per-row descending selection, emit edges + weights -------------
    if (tid < MBLK) {
        const int r    = tid;
        const int arow = rowbase + r;
        for (int ok = 0; ok < KTOP; ++ok) {
            float best = -2.0f; int bk = 0;
#pragma unroll
            for (int k = 0; k < KTOP; ++k) {
                float tv = topv[r][k];
                if (tv > best) { best = tv; bk = k; }
            }
            int bi = topi[r][bk];
            topv[r][bk] = -3.0f;
            int o = arow * KTOP + ok;
            e_src[o] = arow;                       // edges[0, :]
            e_dst[o] = bi;                         // edges[1, :]
            wout[o]  = sqrtf(fmaxf(best, 0.0f));   // sqrt(top-k(sim^2))
        }
    }
}

// ---------------------------------------------------------------------------
extern "C" void kernel_launch(void* const* d_in, const int* in_sizes, int n_in,
                              void* d_out, int out_size, void* d_ws, size_t ws_size,
                              hipStream_t stream) {
    const float* A = (const float*)d_in[0];   // x_actuators [8192,512] fp32
    const float* S = (const float*)d_in[1];   // x_sensors   [8192,512] fp32

    float* inv_na = (float*)d_ws;             // 8192 floats
    float* inv_ns = inv_na + NROWS;           // 8192 floats

    int*   e_src = (int*)d_out;                         // edges row 0
    int*   e_dst = e_src + NROWS * KTOP;                // edges row 1
    float* wts   = (float*)d_out + 2 * NROWS * KTOP;    // weights

    inv_norm_kernel<<<NROWS / 8, 256, 0, stream>>>(A, inv_na);
    inv_norm_kernel<<<NROWS / 8, 256, 0, stream>>>(S, inv_ns);
    simtopk_kernel<<<NROWS / MBLK, 256, 0, stream>>>(A, S, inv_na, inv_ns,
                                                     e_src, e_dst, wts);
}